// PointToMeshResidual_34840774705528
// MI455X (gfx1250) — compile-verified
//
#include <hip/hip_runtime.h>
#include <hip/hip_bf16.h>
#include <stdint.h>

// Point-to-mesh closest query, MI455X (gfx1250, wave32).
// Roofline: ~10 GFLOP fp32 over ~0.5MB inputs -> pure VALU-bound; no GEMM
// structure, so no WMMA. CDNA5 wins used instead: async global->LDS tile
// staging (ASYNCcnt) + division-free inner loop + 8-way triangle split for
// occupancy with packed u64 atomicMin reduction.

#define TPB   256
#define TILE  256
#define NGRP  8

// ---------------- CDNA5 async global->LDS helpers (inline asm) -------------
__device__ __forceinline__ void async_ld_b32(unsigned lds_off, unsigned goff,
                                             const void* sbase) {
  // GVS mode: mem = SGPR64 + VGPR32 offset;  vdst = LDS byte address
  asm volatile("global_load_async_to_lds_b32 %0, %1, %2"
               :: "v"(lds_off), "v"(goff), "s"(sbase) : "memory");
}
__device__ __forceinline__ void wait_async9() {
  asm volatile("s_wait_asynccnt 9" ::: "memory");
}
__device__ __forceinline__ void wait_async0() {
  asm volatile("s_wait_asynccnt 0" ::: "memory");
}

__device__ __forceinline__ float sdiv(float n, float d) {
  return n / (d == 0.0f ? 1.0f : d);          // mirrors reference safe_div
}

// Stage one 256-triangle tile (2304 dwords, flat identity copy) into LDS.
// Each wave issues 9 async-b32 instructions (ASYNCcnt += 9 per wave).
__device__ __forceinline__ void issue_raw_tile(const float* tris, float* rawbuf,
                                               unsigned base_dw, unsigned max_dw,
                                               int tid) {
  unsigned lds0 = (unsigned)(uintptr_t)(rawbuf + tid);
#pragma unroll
  for (int j = 0; j < 9; ++j) {
    unsigned idx = base_dw + (unsigned)j * TILE + (unsigned)tid;
    if (idx >= max_dw) idx = max_dw - 1u;     // clamp: stay in-bounds, dup ok
    async_ld_b32(lds0 + (unsigned)j * (TILE * 4u), idx * 4u, tris);
  }
}

// ---------------- epilogue math (reference-exact branch cascade) -----------
__device__ __forceinline__ void write_result(const float* __restrict__ tris,
                                             const float* __restrict__ pts,
                                             float* __restrict__ out,
                                             int idx, int q, int Q) {
  const float* tv = tris + (size_t)idx * 9;
  float ax=tv[0], ay=tv[1], az=tv[2];
  float bx=tv[3], by=tv[4], bz=tv[5];
  float cx=tv[6], cy=tv[7], cz=tv[8];
  float px=pts[3*q], py=pts[3*q+1], pz=pts[3*q+2];

  float abx=bx-ax, aby=by-ay, abz=bz-az;
  float acx=cx-ax, acy=cy-ay, acz=cz-az;
  float apx=px-ax, apy=py-ay, apz=pz-az;
  float bpx=px-bx, bpy=py-by, bpz=pz-bz;
  float cpx=px-cx, cpy=py-cy, cpz=pz-cz;
  float d1=abx*apx+aby*apy+abz*apz;
  float d2=acx*apx+acy*apy+acz*apz;
  float d3=abx*bpx+aby*bpy+abz*bpz;
  float d4=acx*bpx+acy*bpy+acz*bpz;
  float d5=abx*cpx+aby*cpy+abz*cpz;
  float d6=acx*cpx+acy*cpy+acz*cpz;
  float va=d3*d6-d5*d4, vb=d5*d2-d1*d6, vc=d1*d4-d3*d2;

  float w0, w1, w2;
  if (d1<=0.f && d2<=0.f)                 { w0=1.f; w1=0.f; w2=0.f; }
  else if (d3>=0.f && d4<=d3)             { w0=0.f; w1=1.f; w2=0.f; }
  else if (d6>=0.f && d5<=d6)             { w0=0.f; w1=0.f; w2=1.f; }
  else if (vc<=0.f && d1>=0.f && d3<=0.f) { float v=sdiv(d1, d1-d3);
                                            w0=1.f-v; w1=v;   w2=0.f; }
  else if (vb<=0.f && d2>=0.f && d6<=0.f) { float w=sdiv(d2, d2-d6);
                                            w0=1.f-w; w1=0.f; w2=w;   }
  else if (va<=0.f && (d4-d3)>=0.f && (d5-d6)>=0.f) {
                                            float w=sdiv(d4-d3,(d4-d3)+(d5-d6));
                                            w0=0.f; w1=1.f-w; w2=w;   }
  else { float inv=sdiv(1.f, va+vb+vc);
         float v=vb*inv, w=vc*inv; w0=1.f-v-w; w1=v; w2=w; }

  w0 = fminf(fmaxf(w0, 0.f), 1.f);        // jnp.clip(bcs, 0, 1)
  w1 = fminf(fmaxf(w1, 0.f), 1.f);
  w2 = fminf(fmaxf(w2, 0.f), 1.f);

  float qx = w0*ax + w1*bx + w2*cx;
  float qy = w0*ay + w1*by + w2*cy;
  float qz = w0*az + w1*bz + w2*cz;

  out[3*q+0] = qx - px;
  out[3*q+1] = qy - py;
  out[3*q+2] = qz - pz;
  float* o2 = out + (size_t)Q * 3 + (size_t)q * 9;
  o2[0]=ax; o2[1]=ay; o2[2]=az;
  o2[3]=bx; o2[4]=by; o2[5]=bz;
  o2[6]=cx; o2[7]=cy; o2[8]=cz;
}

// ---------------- core search over a triangle range ------------------------
// Returns packed key (distance bits << 32 | index); scans [t0, t0+cnt).
__device__ __forceinline__ unsigned long long
search_range(const float* __restrict__ tris, int F, int t0, int cnt,
             float px, float py, float pz, int tid,
             float (*raw)[TILE*9], float4* der) {
  const int ntiles = (cnt + TILE - 1) / TILE;
  const unsigned max_dw = (unsigned)F * 9u;

  // prime the double-buffered pipeline
  issue_raw_tile(tris, &raw[0][0], (unsigned)t0 * 9u, max_dw, tid);
  if (ntiles > 1)
    issue_raw_tile(tris, &raw[1][0], ((unsigned)t0 + TILE) * 9u, max_dw, tid);

  float best = 3.402823466e38f;
  int   bidx = t0;

  for (int tk = 0; tk < ntiles; ++tk) {
    if (tk + 1 < ntiles) wait_async9(); else wait_async0(); // tile tk landed
    __syncthreads();   // all waves' async portions in LDS; der free to reuse

    // Per-tile precompute: derived constants incl. reciprocals (kills the
    // 4 transcendental divides per pair the reference formulation needs).
    {
      const float* r = &raw[tk & 1][tid * 9];
      float ax=r[0], ay=r[1], az=r[2];
      float bx=r[3], by=r[4], bz=r[5];
      float cx=r[6], cy=r[7], cz=r[8];
      float abx=bx-ax, aby=by-ay, abz=bz-az;
      float acx=cx-ax, acy=cy-ay, acz=cz-az;
      float bcx=cx-bx, bcy=cy-by, bcz=cz-bz;
      float nx = aby*acz - abz*acy;
      float ny = abz*acx - abx*acz;
      float nz = abx*acy - aby*acx;
      float nab2 = abx*abx + aby*aby + abz*abz;   // == d1-d3
      float nac2 = acx*acx + acy*acy + acz*acz;   // == d2-d6
      float nbc2 = bcx*bcx + bcy*bcy + bcz*bcz;   // == (d4-d3)+(d5-d6)
      float kab  = abx*acx + aby*acy + abz*acz;
      float nn   = nx*nx + ny*ny + nz*nz;
      float iab = (nab2 == 0.f) ? 1.f : 1.f/nab2;
      float iac = (nac2 == 0.f) ? 1.f : 1.f/nac2;
      float ibc = (nbc2 == 0.f) ? 1.f : 1.f/nbc2;
      float inn = (nn   == 0.f) ? 1.f : 1.f/nn;
      der[tid*5+0] = make_float4(ax, ay, az, abx);
      der[tid*5+1] = make_float4(aby, abz, acx, acy);
      der[tid*5+2] = make_float4(acz, nx, ny, nz);
      der[tid*5+3] = make_float4(nab2, nac2, kab, iab);
      der[tid*5+4] = make_float4(iac, ibc, inn, 0.f);
    }
    __syncthreads();   // derived ready; raw[tk&1] fully consumed

    if (tk + 2 < ntiles)   // overlap next-next tile fetch with compute
      issue_raw_tile(tris, &raw[tk & 1][0],
                     ((unsigned)t0 + (unsigned)(tk + 2) * TILE) * 9u,
                     max_dw, tid);

    const int tc = min(TILE, cnt - tk * TILE);
    const int tb = t0 + tk * TILE;
#pragma unroll 2
    for (int t = 0; t < tc; ++t) {
      // uniform-address ds_load_b128 -> broadcast to all 32 lanes
      float4 r0 = der[t*5+0], r1 = der[t*5+1], r2 = der[t*5+2];
      float4 r3 = der[t*5+3], r4 = der[t*5+4];
      float apx = px - r0.x, apy = py - r0.y, apz = pz - r0.z;
      float abx = r0.w, aby = r1.x, abz = r1.y;
      float acx = r1.z, acy = r1.w, acz = r2.x;
      float d1 = fmaf(abx, apx, fmaf(aby, apy, abz * apz));
      float d2 = fmaf(acx, apx, fmaf(acy, apy, acz * apz));
      float d3 = d1 - r3.x;                 // ab.(p-b)
      float d5 = d1 - r3.z;                 // ab.(p-c)
      float d4 = d2 - r3.z;                 // ac.(p-b)
      float d6 = d2 - r3.y;                 // ac.(p-c)
      float va = d3*d6 - d5*d4;
      float vb = d5*d2 - d1*d6;
      float vc = d1*d4 - d3*d2;
      float bpx = apx-abx, bpy = apy-aby, bpz = apz-abz;
      float cqx = apx-acx, cqy = apy-acy, cqz = apz-acz;
      float dap = fmaf(apx,apx, fmaf(apy,apy, apz*apz));
      float dbp = fmaf(bpx,bpx, fmaf(bpy,bpy, bpz*bpz));
      float dcq = fmaf(cqx,cqx, fmaf(cqy,cqy, cqz*cqz));
      float e = d4 - d3, f = d5 - d6;       // bc.bp, -bc.cp
      float tn = fmaf(r2.y,apx, fmaf(r2.z,apy, r2.w*apz));  // n.ap
      // Region distances; override order == reference where-cascade order
      float dist = (tn*tn) * r4.z;                                  // face
      dist = (va<=0.f && e>=0.f && f>=0.f)   ? dbp - (e*e)*r4.y   : dist; // BC
      dist = (vb<=0.f && d2>=0.f && d6<=0.f) ? dap - (d2*d2)*r4.x : dist; // AC
      dist = (vc<=0.f && d1>=0.f && d3<=0.f) ? dap - (d1*d1)*r3.w : dist; // AB
      dist = (d6>=0.f && d5<=d6)             ? dcq : dist;                // C
      dist = (d3>=0.f && d4<=d3)             ? dbp : dist;                // B
      dist = (d1<=0.f && d2<=0.f)            ? dap : dist;                // A
      if (dist < best) { best = dist; bidx = tb + t; }   // first-min tie rule
    }
  }
  // distances >= 0 mathematically; clamp fp noise so uint ordering == float
  unsigned db = __float_as_uint(fmaxf(best, 0.0f));
  return ((unsigned long long)db << 32) | (unsigned)bidx;
}

// ---------------- kernels ---------------------------------------------------
__global__ __launch_bounds__(TPB) void p2m_init(unsigned long long* ws, int Q) {
  int i = blockIdx.x * TPB + threadIdx.x;
  if (i < Q) ws[i] = ~0ull;
}

__global__ __launch_bounds__(TPB)
void p2m_search(const float* __restrict__ tris, const float* __restrict__ pts,
                unsigned long long* __restrict__ ws, int F, int Q) {
  __shared__ float  raw[2][TILE * 9];   // 18 KB raw double buffer
  __shared__ float4 der[TILE * 5];      // 20 KB derived constants
  const int tid = threadIdx.x;
  const int q   = blockIdx.x * TPB + tid;
  const int g   = blockIdx.y;
  const int per = (F + (int)gridDim.y - 1) / (int)gridDim.y;
  const int t0  = g * per;
  const int cnt = min(per, F - t0);
  if (cnt <= 0) return;                 // uniform per block -> barrier-safe

  float px = 0.f, py = 0.f, pz = 0.f;
  if (q < Q) { px = pts[3*q]; py = pts[3*q+1]; pz = pts[3*q+2]; }

  unsigned long long key = search_range(tris, F, t0, cnt, px, py, pz, tid, raw, der);
  if (q < Q) atomicMin(&ws[q], key);    // (dist,idx) lexicographic min
}

__global__ __launch_bounds__(TPB)
void p2m_finish(const float* __restrict__ tris, const float* __restrict__ pts,
                const unsigned long long* __restrict__ ws,
                float* __restrict__ out, int F, int Q) {
  int q = blockIdx.x * TPB + threadIdx.x;
  if (q >= Q) return;
  int idx = (int)(unsigned)(ws[q] & 0xffffffffull);
  if (idx < 0 || idx >= F) idx = 0;     // safety vs poison
  write_result(tris, pts, out, idx, q, Q);
}

// Fallback if the harness workspace is ever too small: one group, fused epilogue.
__global__ __launch_bounds__(TPB)
void p2m_fused(const float* __restrict__ tris, const float* __restrict__ pts,
               float* __restrict__ out, int F, int Q) {
  __shared__ float  raw[2][TILE * 9];
  __shared__ float4 der[TILE * 5];
  const int tid = threadIdx.x;
  const int q   = blockIdx.x * TPB + tid;
  float px = 0.f, py = 0.f, pz = 0.f;
  if (q < Q) { px = pts[3*q]; py = pts[3*q+1]; pz = pts[3*q+2]; }
  unsigned long long key = search_range(tris, F, 0, F, px, py, pz, tid, raw, der);
  if (q < Q) write_result(tris, pts, out, (int)(unsigned)(key & 0xffffffffull), q, Q);
}

extern "C" void kernel_launch(void* const* d_in, const int* in_sizes, int n_in,
                              void* d_out, int out_size, void* d_ws, size_t ws_size,
                              hipStream_t stream) {
  (void)n_in; (void)out_size;
  const float* tris = (const float*)d_in[0];   // [B=1, F, 3, 3] fp32
  const float* pts  = (const float*)d_in[1];   // [B=1, Q, 3] fp32
  float* out = (float*)d_out;                  // residual [Q,3] ++ tris [Q,3,3]
  const int F = in_sizes[0] / 9;               // B==1 in this problem
  const int Q = in_sizes[1] / 3;
  const int pblocks = (Q + TPB - 1) / TPB;

  if (ws_size >= (size_t)Q * sizeof(unsigned long long)) {
    unsigned long long* ws = (unsigned long long*)d_ws;
    p2m_init  <<<pblocks, TPB, 0, stream>>>(ws, Q);
    p2m_search<<<dim3(pblocks, NGRP), TPB, 0, stream>>>(tris, pts, ws, F, Q);
    p2m_finish<<<pblocks, TPB, 0, stream>>>(tris, pts, ws, out, F, Q);
  } else {
    p2m_fused <<<pblocks, TPB, 0, stream>>>(tris, pts, out, F, Q);
  }
}